// LocationSensitiveAttention_57621281243471
// MI455X (gfx1250) — compile-verified
//
#include <hip/hip_runtime.h>
#include <math.h>

#define B_    128
#define T_    1024
#define RNN_  1024
#define EMB_  512
#define ATT_  128
#define NF_   32
#define KS_   31
#define PAD_  15
#define CHUNK_ 256
#define CSTRIDE_ 34          // even stride -> 8B-aligned b64 LDS loads, conflict-free
#define AWS_  1056           // padded aw row (>= T + 2*PAD = 1054, even)

typedef float v2f __attribute__((ext_vector_type(2)));
typedef float v4f __attribute__((ext_vector_type(4)));
typedef float v8f __attribute__((ext_vector_type(8)));

// gfx1250 native v_tanh_f32 transcendental.
__device__ __forceinline__ float fast_tanh(float x) {
#if __has_builtin(__builtin_amdgcn_tanhf)
  return __builtin_amdgcn_tanhf(x);
#elif __has_builtin(__builtin_amdgcn_tanh_f32)
  return __builtin_amdgcn_tanh_f32(x);
#else
  float r;
  asm volatile("v_tanh_f32 %0, %1\n\tv_nop" : "=v"(r) : "v"(x));
  return r;
#endif
}

__launch_bounds__(256)
__global__ void lsa_fused_kernel(const float* __restrict__ hs,      // [B,RNN]
                                 const float* __restrict__ memory,  // [B,T,EMB]
                                 const float* __restrict__ pmem,    // [B,T,ATT]
                                 const float* __restrict__ awcat,   // [B,2,T]
                                 const unsigned char* __restrict__ mask, // [B,T] bool
                                 const float* __restrict__ wq,      // [RNN,ATT]
                                 const float* __restrict__ wconv,   // [NF,2,KS] (OIH)
                                 const float* __restrict__ wloc,    // [NF,ATT]
                                 const float* __restrict__ vvec,    // [ATT]
                                 float* __restrict__ out_ctx,       // [B,EMB]
                                 float* __restrict__ out_w) {       // [B,T]
  __shared__ float s_aw[2 * AWS_];            // 8448 B  (zero-padded halo; == im2col)
  __shared__ float s_conv[CHUNK_ * CSTRIDE_]; // 34816 B (also reused as scratch)
  __shared__ float s_energy[T_];              // 4096 B
  __shared__ float s_pq[ATT_];                // 512 B
  __shared__ float s_red[8];

  const int tid  = threadIdx.x;
  const int b    = blockIdx.x;
  const int lane = tid & 31;
  const int wid  = tid >> 5;
  const int n      = lane & 15;            // tile column
  const int m      = lane & 15;            // tile row (A side)
  const int khalf  = (lane & 16) ? 2 : 0;  // K-pair selector (A/B frags)
  const int rowoff = (lane & 16) ? 8 : 0;  // C/D row offset

  // ---- stage aw (+halo) into LDS; zero energies ----
  for (int i = tid; i < 2 * AWS_; i += 256) {
    int c = i / AWS_, x = i % AWS_;
    int src = x - PAD_;
    float vl = 0.f;
    if (src >= 0 && src < T_) vl = awcat[(b * 2 + c) * T_ + src];
    s_aw[i] = vl;
  }
  for (int t = tid; t < T_; t += 256) s_energy[t] = 0.f;

  // ---- processed_query: pq[a] = sum_r hs[b,r]*wq[r,a]; 2 threads per column ----
  {
    const int a = tid >> 1;
    const int rbase = (tid & 1) * (RNN_ / 2);
    float acc = 0.f;
    #pragma unroll 8
    for (int r = 0; r < RNN_ / 2; ++r)
      acc += hs[b * RNN_ + rbase + r] * wq[(rbase + r) * ATT_ + a];
    s_conv[tid] = acc;            // reuse s_conv as scratch before first chunk
  }
  __syncthreads();
  if (tid < ATT_) s_pq[tid] = s_conv[2 * tid] + s_conv[2 * tid + 1];
  __syncthreads();

  // ---- per-lane constants ----
  const float pq_lane = s_pq[wid * 16 + n];   // a-tile = wid for loc phase
  const float v_lane  = vvec[wid * 16 + n];

  // B fragments for loc GEMM: w_loc[32f x 16a], a-tile = wid (8 waves cover ATT=128)
  const int at = wid;
  v2f bloc[8];
  #pragma unroll
  for (int ks = 0; ks < 8; ++ks) {
    int f0 = 4 * ks + khalf;
    bloc[ks][0] = wloc[f0 * ATT_ + at * 16 + n];
    bloc[ks][1] = wloc[(f0 + 1) * ATT_ + at * 16 + n];
  }

  // B fragments for conv GEMM: W2[64q x 32f]; q = c*32+k, rows q=31,63 are zero pad.
  // Waves 0-3 -> f-tile 0 (f 0..15), waves 4-7 -> f-tile 1 (f 16..31).
  const int ftile = wid >> 2;
  const int fcol  = ftile * 16 + n;
  const int wsub  = wid & 3;
  v2f bconv[16];
  #pragma unroll
  for (int ks = 0; ks < 16; ++ks) {
    #pragma unroll
    for (int e = 0; e < 2; ++e) {
      int q = 4 * ks + khalf + e;
      int c = q >> 5, k = q & 31;
      bconv[ks][e] = (k < KS_) ? wconv[fcol * (2 * KS_) + c * KS_ + k] : 0.f;
    }
  }

  // ---- main loop over T in chunks of 256 ----
  for (int chunk = 0; chunk < T_; chunk += CHUNK_) {
    // Phase 1: conv via WMMA. s_aw is the im2col matrix (sliding window, stride 1):
    // A[t][q] = s_aw[c*AWS + t + (q&31)].  4 t-tiles per wave.
    #pragma unroll
    for (int i = 0; i < 4; ++i) {
      const int tt = wsub * 4 + i;          // 0..15
      const int trow = chunk + tt * 16 + m; // absolute t of this lane's A row
      v8f cacc = {0.f, 0.f, 0.f, 0.f, 0.f, 0.f, 0.f, 0.f};
      #pragma unroll
      for (int ks = 0; ks < 16; ++ks) {
        const int qb = 4 * ks + khalf;      // even; pair never straddles c boundary
        const float* base = &s_aw[(qb >> 5) * AWS_ + trow + (qb & 31)];
        v2f af;
        af[0] = base[0];
        af[1] = base[1];                    // q==31/63 -> garbage * 0 (B row is zero)
        cacc = __builtin_amdgcn_wmma_f32_16x16x4_f32(
            false, af, false, bconv[ks], (short)0, cacc, false, false);
      }
      #pragma unroll
      for (int j = 0; j < 8; ++j)
        s_conv[(tt * 16 + j + rowoff) * CSTRIDE_ + fcol] = cacc[j];
    }
    __syncthreads();

    // Phase 2: loc tile = conv[16t x 32f] @ w_loc[32f x 16a] + fused energy epilogue
    for (int tt = 0; tt < 16; ++tt) {
      const int t0 = chunk + tt * 16;

      const float* abase = &s_conv[(tt * 16 + m) * CSTRIDE_ + khalf];
      v2f afrag[8];
      #pragma unroll
      for (int ks = 0; ks < 8; ++ks)
        afrag[ks] = *(const v2f*)(abase + 4 * ks);

      float pm[8];
      #pragma unroll
      for (int j = 0; j < 8; ++j)
        pm[j] = pmem[(b * T_ + t0 + j + rowoff) * ATT_ + at * 16 + n];

      v8f acc = {0.f, 0.f, 0.f, 0.f, 0.f, 0.f, 0.f, 0.f};
      #pragma unroll
      for (int ks = 0; ks < 8; ++ks)
        acc = __builtin_amdgcn_wmma_f32_16x16x4_f32(
            false, afrag[ks], false, bloc[ks], (short)0, acc, false, false);

      // e[t] += sum_a v[a] * tanh(pq[a] + loc[t,a] + pmem[b,t,a])
      float ev[8];
      #pragma unroll
      for (int j = 0; j < 8; ++j)
        ev[j] = v_lane * fast_tanh(pq_lane + acc[j] + pm[j]);
      // stage-batched 16-lane xor reduction: 8 independent bpermutes per stage
      #pragma unroll
      for (int st = 1; st <= 8; st <<= 1) {
        #pragma unroll
        for (int j = 0; j < 8; ++j)
          ev[j] += __shfl_xor(ev[j], st, 16);
      }
      if (n == 0) {
        #pragma unroll
        for (int j = 0; j < 8; ++j)
          atomicAdd(&s_energy[t0 + j + rowoff], ev[j]);
      }
    }
    __syncthreads();
  }

  // ---- softmax over T (in LDS) ----
  float lmax = -INFINITY;
  for (int t = tid; t < T_; t += 256) {
    float e = s_energy[t];
    if (mask[b * T_ + t]) e = -INFINITY;
    s_energy[t] = e;
    lmax = fmaxf(lmax, e);
  }
  #pragma unroll
  for (int o = 16; o >= 1; o >>= 1) lmax = fmaxf(lmax, __shfl_xor(lmax, o, 32));
  if (lane == 0) s_red[wid] = lmax;
  __syncthreads();
  float bmax = s_red[0];
  #pragma unroll
  for (int i = 1; i < 8; ++i) bmax = fmaxf(bmax, s_red[i]);
  __syncthreads();

  float lsum = 0.f;
  for (int t = tid; t < T_; t += 256) {
    float ex = __expf(s_energy[t] - bmax);
    s_energy[t] = ex;
    lsum += ex;
  }
  #pragma unroll
  for (int o = 16; o >= 1; o >>= 1) lsum += __shfl_xor(lsum, o, 32);
  if (lane == 0) s_red[wid] = lsum;
  __syncthreads();
  float bsum = 0.f;
  #pragma unroll
  for (int i = 0; i < 8; ++i) bsum += s_red[i];
  const float inv = 1.f / bsum;

  for (int t = tid; t < T_; t += 256) {
    float wgt = s_energy[t] * inv;
    s_energy[t] = wgt;
    out_w[b * T_ + t] = wgt;
  }
  __syncthreads();

  // ---- context: c[d] = sum_t w[t] * memory[b,t,d] (bandwidth-dominant stream) ----
  // b128 loads: thread owns a fixed 4-float d-group g, covers t-parity tid>>7.
  {
    const int g = tid & 127;            // d-group: d = 4g .. 4g+3
    const int parity = tid >> 7;        // t ≡ parity (mod 2)
    const v4f* mem4 = (const v4f*)(memory + (size_t)b * T_ * EMB_);
    v4f acc4 = {0.f, 0.f, 0.f, 0.f};
    #pragma unroll 4
    for (int t2 = parity; t2 < T_; t2 += 2) {
      const float wv = s_energy[t2];
      const v4f mm = mem4[t2 * (EMB_ / 4) + g];
      acc4[0] += wv * mm[0];
      acc4[1] += wv * mm[1];
      acc4[2] += wv * mm[2];
      acc4[3] += wv * mm[3];
    }
    // combine the two parity partials through LDS (reuse s_conv as scratch)
    v4f* s_part = (v4f*)s_conv;
    if (parity == 1) s_part[g] = acc4;
    __syncthreads();
    if (parity == 0) {
      const v4f other = s_part[g];
      acc4[0] += other[0];
      acc4[1] += other[1];
      acc4[2] += other[2];
      acc4[3] += other[3];
      *(v4f*)(out_ctx + b * EMB_ + 4 * g) = acc4;
    }
  }
}

extern "C" void kernel_launch(void* const* d_in, const int* in_sizes, int n_in,
                              void* d_out, int out_size, void* d_ws, size_t ws_size,
                              hipStream_t stream) {
  (void)in_sizes; (void)n_in; (void)out_size; (void)d_ws; (void)ws_size;
  const float* hs     = (const float*)d_in[0];
  const float* memory = (const float*)d_in[1];
  const float* pmem   = (const float*)d_in[2];
  const float* awcat  = (const float*)d_in[3];
  const unsigned char* mask = (const unsigned char*)d_in[4];
  const float* wq     = (const float*)d_in[5];
  const float* wconv  = (const float*)d_in[6];
  const float* wloc   = (const float*)d_in[7];
  const float* vvec   = (const float*)d_in[8];

  float* out_ctx = (float*)d_out;                     // [B, EMB] first
  float* out_w   = (float*)d_out + (size_t)B_ * EMB_; // then [B, T]

  lsa_fused_kernel<<<dim3(B_), dim3(256), 0, stream>>>(
      hs, memory, pmem, awcat, mask, wq, wconv, wloc, vvec, out_ctx, out_w);
}